// GraphConvolution_7627861918155
// MI455X (gfx1250) — compile-verified
//
#include <hip/hip_runtime.h>

typedef __attribute__((ext_vector_type(2))) float v2f;
typedef __attribute__((ext_vector_type(8))) float v8f;

#define D_DIM 128
#define O_DIM 128
#define S_REL 8
#define B_BAS 4

// ---------------------------------------------------------------------------
// Kernel 1: basis combine, mirroring the reference's (D,S,O)->(S*D,O) reshape:
//   V[k][o] = sum_b W_comp[k%8][b] * W[b*128 + k/8][o]      k in [0,1024)
// ---------------------------------------------------------------------------
__global__ void rgcn_combine_V(const float* __restrict__ W,
                               const float* __restrict__ Wc,
                               float* __restrict__ V) {
    int idx = blockIdx.x * blockDim.x + threadIdx.x;   // 1024*128 threads
    int o  = idx & (O_DIM - 1);
    int k  = idx >> 7;
    int s2 = k & (S_REL - 1);
    int d2 = k >> 3;
    float acc = 0.f;
#pragma unroll
    for (int b = 0; b < B_BAS; ++b)
        acc += Wc[s2 * B_BAS + b] * W[(b * D_DIM + d2) * O_DIM + o];
    V[k * O_DIM + o] = acc;
}

// ---------------------------------------------------------------------------
// Kernel 2: zero the output accumulator (d_out is poisoned by the harness).
// ---------------------------------------------------------------------------
__global__ void rgcn_zero(float* __restrict__ p, int n) {
    int i = blockIdx.x * blockDim.x + threadIdx.x;
    if (i < n) p[i] = 0.f;
}

// ---------------------------------------------------------------------------
// Kernel 3: dense GEMM  H = F @ Vs   ([N x 128] @ [128 x 128], f32 WMMA).
// One wave computes a 16(M) x 128(O) tile: 8 accumulators of v8f, K-loop in
// steps of 4 using V_WMMA_F32_16X16X4_F32.
//   A 16x4 layout: lanes 0-15 -> {K=k0,k0+1}, lanes 16-31 -> {K=k0+2,k0+3}
//   B 4x16 layout: same K split per lane-half, N striped over lanes 0-15
//   C/D 16x16:     vgpr j, lanes 0-15 -> M=j, lanes 16-31 -> M=8+j
// ---------------------------------------------------------------------------
__global__ __launch_bounds__(256) void rgcn_gemm_fv(const float* __restrict__ F,
                                                    const float* __restrict__ Vs,
                                                    float* __restrict__ H,
                                                    int nTiles) {
    int wave  = threadIdx.x >> 5;
    int mtile = blockIdx.x * 8 + wave;
    if (mtile >= nTiles) return;              // wave-uniform: EXEC stays all-1s
    int lane = threadIdx.x & 31;
    int half = lane >> 4;
    int lm   = lane & 15;

    const float* Arow = F + (size_t)(mtile * 16 + lm) * D_DIM + 2 * half;

    v8f acc[8] = {};

    for (int k0 = 0; k0 < D_DIM; k0 += 4) {
        v2f a;
        a.x = Arow[k0];
        a.y = Arow[k0 + 1];
        const float* Bbase = Vs + (size_t)(k0 + 2 * half) * O_DIM + lm;
#pragma unroll
        for (int t = 0; t < 8; ++t) {
            v2f b;
            b.x = Bbase[t * 16];
            b.y = Bbase[t * 16 + O_DIM];
            acc[t] = __builtin_amdgcn_wmma_f32_16x16x4_f32(
                /*neg_a=*/false, a, /*neg_b=*/false, b,
                /*c_mod=*/(short)0, acc[t],
                /*reuse_a=*/false, /*reuse_b=*/false);
        }
    }

#pragma unroll
    for (int t = 0; t < 8; ++t) {
        float* outp = H + (size_t)(mtile * 16 + half * 8) * O_DIM + t * 16 + lm;
#pragma unroll
        for (int j = 0; j < 8; ++j)
            outp[(size_t)j * O_DIM] = acc[t][j];
    }
}

// ---------------------------------------------------------------------------
// Kernel 4: edge scatter  out[row] += val * H[col]   (one wave per edge,
// 4 floats per lane; f32 hardware atomics -> global_atomic_add_f32).
// ---------------------------------------------------------------------------
__global__ __launch_bounds__(256) void rgcn_spmm_scatter(const int* __restrict__ rows,
                                                         const int* __restrict__ cols,
                                                         const float* __restrict__ vals,
                                                         const float* __restrict__ H,
                                                         float* __restrict__ out,
                                                         int E) {
    int e = (int)((blockIdx.x * blockDim.x + threadIdx.x) >> 5);
    if (e >= E) return;
    int   lane = threadIdx.x & 31;
    int   r = rows[e];
    int   c = cols[e];
    float v = vals[e];
    const float4 h = *(const float4*)(H + (size_t)c * O_DIM + lane * 4);
    float* po = out + (size_t)r * O_DIM + lane * 4;
    unsafeAtomicAdd(po + 0, v * h.x);
    unsafeAtomicAdd(po + 1, v * h.y);
    unsafeAtomicAdd(po + 2, v * h.z);
    unsafeAtomicAdd(po + 3, v * h.w);
}

// ---------------------------------------------------------------------------
// Kernel 5: relu epilogue.
// ---------------------------------------------------------------------------
__global__ void rgcn_relu(float* __restrict__ p, int n) {
    int i = blockIdx.x * blockDim.x + threadIdx.x;
    if (i < n) p[i] = fmaxf(p[i], 0.f);
}

// ---------------------------------------------------------------------------
// Host-side launcher.
// Inputs (setup_inputs order): features[N*128], rows[S*E], cols[S*E],
// vals[S*E], W[512*128], W_comp[8*4].   Output: relu(out)[N*128] f32.
// Workspace: Vmat [1024*128] f32 (512 KB) + H [N*128] f32 (~51.2 MB).
// ---------------------------------------------------------------------------
extern "C" void kernel_launch(void* const* d_in, const int* in_sizes, int n_in,
                              void* d_out, int out_size, void* d_ws, size_t ws_size,
                              hipStream_t stream) {
    const float* features = (const float*)d_in[0];
    const int*   rows     = (const int*)d_in[1];
    const int*   cols     = (const int*)d_in[2];
    const float* vals     = (const float*)d_in[3];
    const float* W        = (const float*)d_in[4];
    const float* Wc       = (const float*)d_in[5];
    float*       out      = (float*)d_out;

    const int N = in_sizes[0] / D_DIM;     // 100000
    const int E = in_sizes[1] / S_REL;     // 1600000

    float* Vmat = (float*)d_ws;                                        // [1024][128]
    float* H    = (float*)((char*)d_ws + (size_t)1024 * O_DIM * sizeof(float)); // [N][128]

    // 1) combine basis weights
    rgcn_combine_V<<<(1024 * O_DIM) / 256, 256, 0, stream>>>(W, Wc, Vmat);

    // 2) zero output accumulator
    const int no = N * O_DIM;
    rgcn_zero<<<(no + 255) / 256, 256, 0, stream>>>(out, no);

    // 3) per-relation: dense WMMA GEMM, then edge scatter (stream-ordered)
    const int nTiles     = (N + 15) / 16;            // 6250 (N divisible by 16)
    const int gemmBlocks = (nTiles + 7) / 8;
    const int spmmBlocks = (int)(((long long)E * 32 + 255) / 256);
    for (int s = 0; s < S_REL; ++s) {
        rgcn_gemm_fv<<<gemmBlocks, 256, 0, stream>>>(
            features, Vmat + (size_t)s * D_DIM * O_DIM, H, nTiles);
        rgcn_spmm_scatter<<<spmmBlocks, 256, 0, stream>>>(
            rows + (size_t)s * E, cols + (size_t)s * E, vals + (size_t)s * E,
            H, out, E);
    }

    // 4) relu epilogue
    rgcn_relu<<<(no + 255) / 256, 256, 0, stream>>>(out, no);
}